// HardtargetsLoss_36704790511661
// MI455X (gfx1250) — compile-verified
//
#include <hip/hip_runtime.h>

#define T_LEN  2048
#define BATCH  32
#define VOCAB  512
#define TGT    256
#define S_EXT  513
#define NEGINF (-1e30f)

typedef float v2f __attribute__((ext_vector_type(2)));
typedef float v8f __attribute__((ext_vector_type(8)));

// ---------------------------------------------------------------------------
// Kernel 1: logZ[t,b] = max_v x + log(sum_v exp(x - max)), one wave per row.
// Sum-of-exp reduction runs on the WMMA pipe: A = ones(16x4), B = exp chunk,
// C accumulates over 8 wmmas (64 values each -> 512). With A == 1 everywhere,
// D[m,n] = sum_k B[k,n] independent of the B VGPR layout, so summing C's
// first VGPR across the 16-lane group gives the exact row sum.
// ---------------------------------------------------------------------------
__global__ __launch_bounds__(256) void logz_kernel(const float* __restrict__ x,
                                                   float* __restrict__ logZ) {
    const int wid  = threadIdx.x >> 5;
    const int lane = threadIdx.x & 31;
    const int row  = blockIdx.x * 8 + wid;          // row = t*BATCH + b, < 65536
    const float* rp = x + (size_t)row * VOCAB;

    float4 d0 = *(const float4*)(rp + 0   + lane * 4);
    float4 d1 = *(const float4*)(rp + 128 + lane * 4);
    float4 d2 = *(const float4*)(rp + 256 + lane * 4);
    float4 d3 = *(const float4*)(rp + 384 + lane * 4);

    float v[16] = {d0.x, d0.y, d0.z, d0.w, d1.x, d1.y, d1.z, d1.w,
                   d2.x, d2.y, d2.z, d2.w, d3.x, d3.y, d3.z, d3.w};

    float m = v[0];
#pragma unroll
    for (int j = 1; j < 16; ++j) m = fmaxf(m, v[j]);
#pragma unroll
    for (int off = 16; off >= 1; off >>= 1) m = fmaxf(m, __shfl_xor(m, off, 32));

    float ex[16];
#pragma unroll
    for (int j = 0; j < 16; ++j) ex[j] = expf(v[j] - m);

    v2f ones = {1.0f, 1.0f};
    v8f c = {0.f, 0.f, 0.f, 0.f, 0.f, 0.f, 0.f, 0.f};
#pragma unroll
    for (int j = 0; j < 8; ++j) {
        v2f bv = {ex[2 * j], ex[2 * j + 1]};
        c = __builtin_amdgcn_wmma_f32_16x16x4_f32(
                /*neg_a=*/false, ones, /*neg_b=*/false, bv,
                /*c_mod=*/(short)0, c, /*reuse_a=*/false, /*reuse_b=*/false);
    }

    // C VGPR0: lanes 0..15 hold column sums N=0..15 (rows identical); reduce 16.
    float s = c[0];
#pragma unroll
    for (int off = 8; off >= 1; off >>= 1) s += __shfl_xor(s, off, 32);

    float lz = m + logf(s);
    if (lane == 0) {
        int t = row / BATCH, b = row % BATCH;
        logZ[b * T_LEN + t] = lz;                   // transposed for DP locality
    }
}

// ---------------------------------------------------------------------------
// Kernel 2: banded Viterbi DP, one block per batch sample, state-per-thread.
// buf[.][i+2] = {alpha, g}; slots 0,1 hold {NEGINF, 0} so alpha[i-1]/[i-2]
// reads need no branches. g[i] = exact backtrace emission sum (replicates the
// reference's out-of-band ptr=0 behavior), eliminating pointer storage and
// the dependent backtrace chain entirely.
// ---------------------------------------------------------------------------
__global__ __launch_bounds__(544) void viterbi_kernel(const float* __restrict__ x,
                                                      const int* __restrict__ tgts,
                                                      const float* __restrict__ logZ,
                                                      float* __restrict__ xsum) {
    const int b   = blockIdx.x;
    const int tid = threadIdx.x;
    const bool act = tid < S_EXT;

    __shared__ float2 buf[2][S_EXT + 2];
    __shared__ float  lzrow[T_LEN];

    for (int k = tid; k < T_LEN; k += 544) lzrow[k] = logZ[b * T_LEN + k];

    int  lab = 0;
    bool skip_ok = false;
    if (act && (tid & 1)) {
        int j = tid >> 1;
        lab = tgts[b * TGT + j];
        skip_ok = (j >= 1) && (tgts[b * TGT + j - 1] != lab);
    }
    if (tid < 2) {
        buf[0][tid] = make_float2(NEGINF, 0.f);
        buf[1][tid] = make_float2(NEGINF, 0.f);
    }

    // ---- t = 0 init: alpha0 = {logp[0,ext0], logp[0,ext1], 0, ...}, g0 = x0 ----
    float e = 0.f;
    if (act) e = x[b * VOCAB + lab];                 // row t=0
    float lz0 = logZ[b * T_LEN];
    if (act) {
        float a0 = (tid < 2) ? (e - lz0) : 0.f;
        buf[0][tid + 2] = make_float2(a0, e);
    }
    float e_pref = 0.f;
    if (act) e_pref = x[((size_t)1 * BATCH + b) * VOCAB + lab];   // t=1 prefetch
    __syncthreads();

    int cur = 0;
    for (int k = 1; k < T_LEN; ++k) {
        float ecur = e_pref;
        int kn = (k + 1 < T_LEN) ? (k + 1) : k;
        if (act) e_pref = x[((size_t)kn * BATCH + b) * VOCAB + lab];  // hide latency

        if (act) {
            float2 c0 = buf[cur][tid + 2];   // alpha/g at i
            float2 c1 = buf[cur][tid + 1];   // at i-1 (slot 1 = NEGINF for i==0)
            float2 c2 = buf[cur][tid];       // at i-2
            float2 cz = buf[cur][2];         // state 0 (broadcast)

            float a  = c0.x;
            float f1 = c1.x;
            float f2 = skip_ok ? c2.x : NEGINF;

            // reference tie-break: stay, then i-1, then i-2 (strict >)
            float v1 = (f1 > a)  ? f1 : a;   int s1  = (f1 > a)  ? 1 : 0;
            float v2 = (f2 > v1) ? f2 : v1;  int sel = (f2 > v1) ? 2 : s1;

            int start = S_EXT - 2 * (T_LEN - k); if (start < 0) start = 0;
            int end   = 2 * k + 2;               if (end > S_EXT) end = S_EXT;
            bool ib   = (tid >= start) && (tid < end);

            float gsel = (sel == 0) ? c0.y : ((sel == 1) ? c1.y : c2.y);
            float anew = ib ? (v2 + (ecur - lzrow[k])) : a;
            float gnew = ecur + (ib ? gsel : cz.y);   // out-of-band ptr -> state 0

            buf[cur ^ 1][tid + 2] = make_float2(anew, gnew);
        }
        __syncthreads();
        cur ^= 1;
    }

    if (tid == 0) {
        float aS2 = buf[cur][S_EXT].x;       // alpha[S-2]
        float aS1 = buf[cur][S_EXT + 1].x;   // alpha[S-1]
        int i0 = (aS2 > aS1) ? (S_EXT - 2) : (S_EXT - 1);
        xsum[b] = buf[cur][i0 + 2].y;        // sum_t x[t, label_t] for this sample
    }
}

// ---------------------------------------------------------------------------
// Kernel 3: loss = (sum_{t,b} logZ - sum_b xsum[b]) / L. Deterministic tree
// reduce (no float atomics -> bit-stable across graph replays).
// ---------------------------------------------------------------------------
__global__ __launch_bounds__(1024) void finalize_kernel(const float* __restrict__ logZ,
                                                        const float* __restrict__ xsum,
                                                        float* __restrict__ out) {
    __shared__ float red[1024];
    float s = 0.f;
    for (int i = threadIdx.x; i < T_LEN * BATCH; i += 1024) s += logZ[i];
    red[threadIdx.x] = s;
    __syncthreads();
#pragma unroll
    for (int off = 512; off >= 1; off >>= 1) {
        if (threadIdx.x < off) red[threadIdx.x] += red[threadIdx.x + off];
        __syncthreads();
    }
    if (threadIdx.x == 0) {
        float xs = 0.f;
        for (int b2 = 0; b2 < BATCH; ++b2) xs += xsum[b2];
        out[0] = (red[0] - xs) * (1.0f / (float)TGT);
    }
}

extern "C" void kernel_launch(void* const* d_in, const int* in_sizes, int n_in,
                              void* d_out, int out_size, void* d_ws, size_t ws_size,
                              hipStream_t stream) {
    const float* x    = (const float*)d_in[0];   // [T, B, V] f32
    const int*   tgts = (const int*)d_in[1];     // [B*L] int32
    (void)in_sizes; (void)n_in; (void)out_size; (void)ws_size;

    float* ws_logZ = (float*)d_ws;               // 65536 floats
    float* ws_xsum = ws_logZ + T_LEN * BATCH;    // 32 floats

    logz_kernel<<<(T_LEN * BATCH) / 8, 256, 0, stream>>>(x, ws_logZ);
    viterbi_kernel<<<BATCH, 544, 0, stream>>>(x, tgts, ws_logZ, ws_xsum);
    finalize_kernel<<<1, 1024, 0, stream>>>(ws_logZ, ws_xsum, (float*)d_out);
}